// FreCalc_5643587027144
// MI455X (gfx1250) — compile-verified
//
#include <hip/hip_runtime.h>
#include <math.h>

#define NLATg 128
#define NLONg 256
#define LMAXg 50
#define MMAXg 50
#define NPTS  1024
#define NBAT  2

typedef __attribute__((ext_vector_type(2))) float v2f;
typedef __attribute__((ext_vector_type(8))) float v8f;

// workspace layout (float units)
#define OFF_PTS    0         // 2*1024 float4 = 8192 floats {s0,s1,ss,r}
#define OFF_WCC    8192      // 128 doubles (256 float slots)
#define OFF_W      8448      // 50*50*128 = 320000
#define OFF_INTERP 328448    // 2*128*256 = 65536
#define OFF_XR     393984    // 64*256 = 16384   (xr[m][row], row=b*128+k)
#define OFF_COS    410368    // 256*64 = 16384   (cos[j][m])

#ifndef M_PI
#define M_PI 3.14159265358979323846
#endif

// ---------------- Stage 1: cartesian -> spherical (packed float4) ---------
__global__ void k_sph(const float* __restrict__ tgt, float* __restrict__ ws) {
    int t = blockIdx.x * blockDim.x + threadIdx.x;   // 0..2047 = b*1024+n
    if (t >= NBAT * NPTS) return;
    float x = tgt[t * 3 + 0], y = tgt[t * 3 + 1], z = tgt[t * 3 + 2];
    float r   = sqrtf(x * x + y * y + z * z);
    float ryz = sqrtf(y * y + z * z);
    float phi1 = acosf(x / r);
    float a    = acosf(y / ryz);
    float phi2 = (z < 0.0f) ? (2.0f * (float)M_PI - a) : a;
    float s0 = phi1, s1 = phi2 - (float)M_PI;
    float4 p = make_float4(s0, s1, s0 * s0 + s1 * s1, r);
    ((float4*)(ws + OFF_PTS))[t] = p;
}

// ---------------- Stage 2: Clenshaw-Curtis weights (127-pt real iDFT) -----
__device__ __forceinline__ double cc_vold(int t) {
    // v_old over length-128: t<63: 2/((2t+1)(2t-1)); t==63: 1/125; else 0
    if (t < 63) { double n = 2.0 * t + 1.0; return 2.0 / (n * (n - 2.0)); }
    if (t == 63) return 1.0 / 125.0;
    return 0.0;
}
__global__ void k_wcc(float* __restrict__ ws) {
    __shared__ double u[127];
    int i = threadIdx.x;
    if (i < 127) {
        double g = -1.0;
        if (i == 63 || i == 64) g += 127.0;
        g /= (127.0 * 127.0);
        u[i] = -(cc_vold(i) + cc_vold(127 - i)) + g;
    }
    __syncthreads();
    int j = (i == 127) ? 0 : i;   // wcc[127] = wcc[0]
    double s = 0.0;
    for (int k = 0; k < 127; ++k)
        s += u[k] * cos(2.0 * M_PI * (double)k * (double)j / 127.0);
    ((double*)(ws + OFF_WCC))[i] = s / 127.0;
}

// ---------------- Stage 3: Legendre weights W[m][l][k] (f64 recurrence) ---
__global__ void k_legw(float* __restrict__ ws) {
    int k = threadIdx.x;                 // 0..127 latitude node
    if (k >= NLATg) return;
    double x  = cos((double)k * M_PI / 127.0);
    double wk = ((const double*)(ws + OFF_WCC))[k];
    float* W = ws + OFF_W;               // [(m*50+l)*128 + k]
    double diag = 1.0 / sqrt(4.0 * M_PI);          // vdm[0,0]
    for (int m = 0; m < MMAXg; ++m) {
        double sgn = (m & 1) ? -1.0 : 1.0;
        for (int l = 0; l < m; ++l) W[(m * LMAXg + l) * NLATg + k] = 0.0f;
        double pm2 = diag;                          // vdm[m, m]
        W[(m * LMAXg + m) * NLATg + k] = (float)(sgn * pm2 * wk);
        if (m + 1 < LMAXg) {
            double pm1 = sqrt(2.0 * m + 3.0) * x * pm2;   // vdm[m, m+1]
            W[(m * LMAXg + m + 1) * NLATg + k] = (float)(sgn * pm1 * wk);
            for (int l = m + 2; l < LMAXg; ++l) {
                double dl = (double)l, dm = (double)m;
                double c1 = sqrt((2.0 * dl - 1.0) / (dl - dm) * (2.0 * dl + 1.0) / (dl + dm));
                double c2 = sqrt((dl + dm - 1.0) / (dl - dm) * (2.0 * dl + 1.0) /
                                 (2.0 * dl - 3.0) * (dl - dm - 1.0) / (dl + dm));
                double p = x * c1 * pm1 - c2 * pm2;
                W[(m * LMAXg + l) * NLATg + k] = (float)(sgn * p * wk);
                pm2 = pm1; pm1 = p;
            }
        }
        // advance diagonal: vdm[m+1,m+1]
        diag = sqrt((2.0 * (m + 1) + 1.0) * (1.0 + x) * (1.0 - x) / (2.0 * (m + 1))) * diag;
    }
}

// ---------------- Stage 4: cosine-transform matrix C[j][m] ----------------
__global__ void k_cos(float* __restrict__ ws) {
    int t = blockIdx.x * blockDim.x + threadIdx.x;   // 0..16383 = j*64+m
    if (t >= NLONg * 64) return;
    int j = t >> 6, m = t & 63;
    double v = cos(2.0 * M_PI * (double)m * (double)j / (double)NLONg) *
               (2.0 * M_PI / (double)NLONg);
    ws[OFF_COS + t] = (float)v;
}

// ---------------- Stage 5: fused distance + top-3 + interpolation ---------
// 256 blocks = (b,i) rows; 256 threads = longitude j. Points staged in LDS
// as float4 {s0,s1,ss,r} -> one ds_load_b128 (broadcast) per point.
__global__ void k_interp(float* __restrict__ ws) {
    __shared__ float4 pts[NPTS];
    int b = blockIdx.x / NLATg;
    int i = blockIdx.x % NLATg;
    int j = threadIdx.x;
    const float4* gp = (const float4*)(ws + OFF_PTS) + b * NPTS;
    for (int n = threadIdx.x; n < NPTS; n += blockDim.x)
        pts[n] = gp[n];
    __syncthreads();

    float g0 = (float)i * ((float)M_PI / (float)NLATg);
    float g1 = (float)(j - NLATg) * ((float)M_PI / (float)NLATg);
    float gg = g0 * g0 + g1 * g1;
    float a0 = -2.0f * g0, a1 = -2.0f * g1;

    float t0 = 3.4e38f, t1 = 3.4e38f, t2 = 3.4e38f;
    int   i0 = 0, i1 = 0, i2 = 0;
    #pragma unroll 4
    for (int n = 0; n < NPTS; ++n) {
        float4 p = pts[n];
        float d = fmaf(a0, p.x, fmaf(a1, p.y, gg + p.z));
        if (d < t2) {
            if (d < t1) {
                t2 = t1; i2 = i1;
                if (d < t0) { t1 = t0; i1 = i0; t0 = d; i0 = n; }
                else        { t1 = d;  i1 = n; }
            } else { t2 = d; i2 = n; }
        }
    }
    float d0 = sqrtf(fmaxf(t0, 0.0f));
    float d1 = sqrtf(fmaxf(t1, 0.0f));
    float d2 = sqrtf(fmaxf(t2, 0.0f));
    float inv = 1.0f / (d0 + d1 + d2);
    float val = (pts[i0].w * d0 + pts[i1].w * d1 + pts[i2].w * d2) * inv;
    // row = b*128 + i, col = j  (matches reshape (-1,128,256))
    ws[OFF_INTERP + (b * NLATg + i) * NLONg + j] = val;
}

// ---------------- Stage 6: rfft real part as WMMA f32 GEMM ----------------
// XR[m][row] = sum_j INTERP[row][j] * COS[j][m],  row = b*128+k
// M=256 rows (16 tiles), N=64 cols (4 tiles), K=256 (64 steps of 4).
__global__ void k_fft_wmma(float* __restrict__ ws) {
    const float* A = ws + OFF_INTERP;   // [row*256 + j]
    const float* B = ws + OFF_COS;      // [j*64 + m]
    float* XR = ws + OFF_XR;            // [m*256 + row]

    int lane = threadIdx.x;             // wave32, one wave per block
    int tile = blockIdx.x;              // 0..63
    int M0 = (tile & 15) * 16;
    int N0 = (tile >> 4) * 16;
    int lm = lane & 15;                 // M (for A) / N (for B) within tile
    int kh = (lane >> 4) * 2;           // K sub-offset: 0 or 2

    v8f c = {};
    for (int k0 = 0; k0 < NLONg; k0 += 4) {
        v2f a, bm;
        a.x  = A[(M0 + lm) * NLONg + k0 + kh + 0];
        a.y  = A[(M0 + lm) * NLONg + k0 + kh + 1];
        bm.x = B[(k0 + kh + 0) * 64 + N0 + lm];
        bm.y = B[(k0 + kh + 1) * 64 + N0 + lm];
        c = __builtin_amdgcn_wmma_f32_16x16x4_f32(
                false, a, false, bm, (short)0, c, false, false);
    }
    int m = N0 + lm;
    if (m < MMAXg) {
        int rbase = M0 + (lane >> 4) * 8;     // lanes 0-15: M=v, 16-31: M=v+8
        #pragma unroll
        for (int v = 0; v < 8; ++v)
            XR[m * (NBAT * NLATg) + rbase + v] = c[v];
    }
}

// ---------------- Stage 7: out[b,l,m] = sum_k XR[m][b*128+k] * W[m][l][k] -
__global__ void k_contract(const float* __restrict__ ws, float* __restrict__ out) {
    int t = blockIdx.x * blockDim.x + threadIdx.x;   // 0..4999
    if (t >= NBAT * LMAXg * MMAXg) return;
    int b = t / (LMAXg * MMAXg);
    int l = (t / MMAXg) % LMAXg;
    int m = t % MMAXg;
    const float* xr = ws + OFF_XR + m * (NBAT * NLATg) + b * NLATg;
    const float* w  = ws + OFF_W  + (m * LMAXg + l) * NLATg;
    float acc = 0.0f;
    #pragma unroll 4
    for (int k = 0; k < NLATg; ++k) acc = fmaf(xr[k], w[k], acc);
    out[t] = acc;   // flat (b,l,m) row-major
}

extern "C" void kernel_launch(void* const* d_in, const int* in_sizes, int n_in,
                              void* d_out, int out_size, void* d_ws, size_t ws_size,
                              hipStream_t stream) {
    const float* tgt = (const float*)d_in[0];
    float* ws  = (float*)d_ws;
    float* out = (float*)d_out;

    k_sph     <<<8,   256, 0, stream>>>(tgt, ws);
    k_wcc     <<<1,   128, 0, stream>>>(ws);
    k_legw    <<<1,   128, 0, stream>>>(ws);
    k_cos     <<<64,  256, 0, stream>>>(ws);
    k_interp  <<<NBAT * NLATg, NLONg, 0, stream>>>(ws);
    k_fft_wmma<<<64,  32,  0, stream>>>(ws);
    k_contract<<<20,  256, 0, stream>>>(ws, out);
}